// PairDistanceLoss_84061099917860
// MI455X (gfx1250) — compile-verified
//
#include <hip/hip_runtime.h>

// PairDistanceLoss for MI455X (gfx1250, wave32).
// N=16384 rows, C=4096 cols (from the reference setup).
// Stage 1: one wave = 16 rows. Row columns split into 4 chunks of 1024;
//          per-chunk partial sums land exactly in the V_WMMA_F32_16X16X4_F32
//          A-matrix layout, and one f32 WMMA vs an all-ones B reduces K=4 and
//          transposes row sums into the D layout (lanes 0/16 hold 8 rows each).
// Stage 2: deterministic single-block tree reduction of 16384 row losses.

typedef __attribute__((ext_vector_type(2))) float v2f;
typedef __attribute__((ext_vector_type(8))) float v8f;
typedef __attribute__((ext_vector_type(4))) float f4;
typedef __attribute__((ext_vector_type(4))) int   i4;

#define NROWS 16384
#define NCOLS 4096
#define CHUNK (NCOLS / 4)      // 1024 columns per K-slot
#define WAVES_PER_BLOCK 8
#define ROWS_PER_BLOCK (16 * WAVES_PER_BLOCK)

__global__ __launch_bounds__(256) void
pairloss_rows_kernel(const float* __restrict__ x,
                     const int*   __restrict__ y,
                     float*       __restrict__ rowloss) {
    const int wave  = threadIdx.x >> 5;
    const int lane  = threadIdx.x & 31;
    const int g     = blockIdx.x * WAVES_PER_BLOCK + wave;  // 16-row group id
    const int mrow  = lane & 15;                            // row within group
    const int khalf = lane >> 4;                            // 0: K=0,1  1: K=2,3

    const long long rowBase = (long long)(g * 16 + mrow) * NCOLS;

    float accP[2], accN[2], accC[2];
#pragma unroll
    for (int h = 0; h < 2; ++h) {
        const int c = 2 * khalf + h;                        // K-slot 0..3
        const float* xp = x + rowBase + (long long)c * CHUNK;
        const int*   yp = y + rowBase + (long long)c * CHUNK;
        float p = 0.0f, q = 0.0f, n = 0.0f;
        for (int j = 0; j < CHUNK; j += 4) {
            f4 xv = __builtin_nontemporal_load((const f4*)(xp + j));
            i4 yv = __builtin_nontemporal_load((const i4*)(yp + j));
#pragma unroll
            for (int t = 0; t < 4; ++t) {
                const bool  m = (yv[t] == 1);
                const float e = __expf(m ? xv[t] : -xv[t]); // exp(x) or exp(-x)
                p += m ? e    : 0.0f;
                q += m ? 0.0f : e;
                n += m ? 1.0f : 0.0f;
            }
        }
        accP[h] = p; accN[h] = q; accC[h] = n;
    }

    // A-matrix (16x4 f32): lanes 0-15 -> {K0,K1}, lanes 16-31 -> {K2,K3}.
    v2f aP, aN, aC, ones;
    aP[0] = accP[0]; aP[1] = accP[1];
    aN[0] = accN[0]; aN[1] = accN[1];
    aC[0] = accC[0]; aC[1] = accC[1];
    ones[0] = 1.0f;  ones[1] = 1.0f;   // B = 4x16 all-ones (layout-invariant)
    v8f z = {};

    // D[m][n] = sum_k A[m][k] for every column n -> row sums, transposed into
    // the C/D layout (VGPR v: lanes 0-15 hold M=v, lanes 16-31 hold M=v+8).
    v8f dP = __builtin_amdgcn_wmma_f32_16x16x4_f32(false, aP, false, ones,
                                                   (short)0, z, false, false);
    v8f dN = __builtin_amdgcn_wmma_f32_16x16x4_f32(false, aN, false, ones,
                                                   (short)0, z, false, false);
    v8f dC = __builtin_amdgcn_wmma_f32_16x16x4_f32(false, aC, false, ones,
                                                   (short)0, z, false, false);

    // Lane 0 holds rows 0..7 (N=0 column), lane 16 holds rows 8..15.
    if (mrow == 0) {
        const int base = g * 16 + khalf * 8;
#pragma unroll
        for (int v = 0; v < 8; ++v) {
            const float pp = dP[v];
            const float qq = dN[v];
            const float nn = dC[v];
            const float denom = nn * ((float)NCOLS - nn);
            rowloss[base + v] = pp * qq / denom;
        }
    }
}

__global__ __launch_bounds__(256) void
pairloss_reduce_kernel(const float* __restrict__ rowloss,
                       float*       __restrict__ out) {
    __shared__ float sm[256];
    float s = 0.0f;
    for (int i = threadIdx.x; i < NROWS; i += 256) s += rowloss[i];
    sm[threadIdx.x] = s;
    __syncthreads();
#pragma unroll
    for (int off = 128; off > 0; off >>= 1) {
        if ((int)threadIdx.x < off) sm[threadIdx.x] += sm[threadIdx.x + off];
        __syncthreads();
    }
    if (threadIdx.x == 0) out[0] = sm[0] / (float)NROWS;
}

extern "C" void kernel_launch(void* const* d_in, const int* in_sizes, int n_in,
                              void* d_out, int out_size, void* d_ws, size_t ws_size,
                              hipStream_t stream) {
    (void)in_sizes; (void)n_in; (void)out_size; (void)ws_size;
    const float* x = (const float*)d_in[0];
    const int*   y = (const int*)d_in[1];      // 0/1 labels (int32 per harness)
    float* rowloss = (float*)d_ws;             // 16384 floats of scratch

    pairloss_rows_kernel<<<NROWS / ROWS_PER_BLOCK, 256, 0, stream>>>(x, y, rowloss);
    pairloss_reduce_kernel<<<1, 256, 0, stream>>>(rowloss, (float*)d_out);
}